// HierarchicalSoftmax_3298534884000
// MI455X (gfx1250) — compile-verified
//
#include <hip/hip_runtime.h>
#include <hip/hip_bf16.h>
#include <cstdint>
#include <cstddef>

#define VOCAB 50000
#define DEPTH 18
#define HID   256
#define BATCH 16384

typedef __attribute__((ext_vector_type(2))) float v2f;
typedef __attribute__((ext_vector_type(8))) float v8f;

// stable softplus(x) - x*t  (BCE-with-logits core)
__device__ __forceinline__ float bce_term(float x, float t) {
    return fmaxf(x, 0.0f) + log1pf(__expf(-fabsf(x))) - x * t;
}

// One wave = one sample. WMMA f32 16x16x4 computes logits d=0..15,
// VALU handles d=16,17. Block (128 thr = 4 waves) emits one partial sum.
__global__ __launch_bounds__(128) void hs_loss_stage1(
    const float* __restrict__ hidden,
    const int*   __restrict__ target_words,
    const float* __restrict__ W,
    const float* __restrict__ path_bits,
    const int*   __restrict__ path_len,
    float*       __restrict__ block_partials)
{
    __shared__ float hidLds[4][HID];     // per-wave hidden row
    __shared__ float wavePartial[4];

    const int lane   = threadIdx.x & 31;
    const int wave   = threadIdx.x >> 5;
    const int sample = blockIdx.x * 4 + wave;   // 4096 * 4 == BATCH exactly

    // ---- stage this wave's hidden row into LDS (2 x float4 per lane) ----
    const float4* hsrc = (const float4*)(hidden + (size_t)sample * HID);
    float4* hdst = (float4*)hidLds[wave];
    hdst[lane]      = hsrc[lane];
    hdst[lane + 32] = hsrc[lane + 32];

    // wave-uniform gather metadata -> scalar regs / K$ path
    const int w   = __builtin_amdgcn_readfirstlane(target_words[sample]);
    const int len = __builtin_amdgcn_readfirstlane(path_len[w]);
    const float* wrow = W + (size_t)w * DEPTH * HID;
    const float* bitw = path_bits + (size_t)w * DEPTH;

    // ---- WMMA chain over K=256 in steps of 4 ----
    // A layout (16x4 f32): lanes 0-15 hold M=lane, K = 4k+{0,1};
    //                      lanes 16-31 hold M=lane-16, K = 4k+{2,3}.
    // B layout (4x16 f32): VGPR0 = row K0 (low half) / K2 (high half),
    //                      VGPR1 = row K1 / K3; broadcast across N columns.
    const int   koff  = (lane < 16) ? 0 : 2;
    const float* aptr = wrow + (size_t)(lane & 15) * HID + koff;
    const float* bptr = hidLds[wave] + koff;

    v8f acc = {};
    #pragma unroll 8
    for (int k = 0; k < 64; ++k) {
        v2f a = *(const v2f*)(aptr + 4 * k);   // global b64, each lane streams its row
        v2f b = *(const v2f*)(bptr + 4 * k);   // ds_load_b64 from LDS
        acc = __builtin_amdgcn_wmma_f32_16x16x4_f32(
            /*neg_a=*/false, a, /*neg_b=*/false, b,
            /*c_mod=*/(short)0, acc, /*reuse_a=*/false, /*reuse_b=*/false);
    }

    // ---- BCE over the 16 WMMA depths ----
    // All D columns are identical: low-half lanes hold logits[0..7] in acc[0..7],
    // high-half lanes hold logits[8..15].
    const int dbase = (lane < 16) ? 0 : 8;
    float part = 0.0f;
    #pragma unroll
    for (int r = 0; r < 8; ++r) {
        const int d = dbase + r;
        const float m = (d < len) ? 1.0f : 0.0f;
        part += m * bce_term(acc[r], bitw[d]);
    }
    part += __shfl_xor(part, 16);   // low half (d0-7) + high half (d8-15)

    // ---- depths 16,17 via VALU dot + wave reduction ----
    const float4* hl  = (const float4*)hidLds[wave];
    const float4* r16 = (const float4*)(wrow + 16 * HID);
    const float4* r17 = (const float4*)(wrow + 17 * HID);
    float4 h0 = hl[lane * 2], h1 = hl[lane * 2 + 1];
    float4 a0 = r16[lane * 2], a1 = r16[lane * 2 + 1];
    float4 c0 = r17[lane * 2], c1 = r17[lane * 2 + 1];
    float p16 = h0.x*a0.x + h0.y*a0.y + h0.z*a0.z + h0.w*a0.w
              + h1.x*a1.x + h1.y*a1.y + h1.z*a1.z + h1.w*a1.w;
    float p17 = h0.x*c0.x + h0.y*c0.y + h0.z*c0.z + h0.w*c0.w
              + h1.x*c1.x + h1.y*c1.y + h1.z*c1.z + h1.w*c1.w;
    #pragma unroll
    for (int off = 16; off >= 1; off >>= 1) {
        p16 += __shfl_xor(p16, off);
        p17 += __shfl_xor(p17, off);
    }
    if (16 < len) part += bce_term(p16, bitw[16]);
    if (17 < len) part += bce_term(p17, bitw[17]);

    part /= (float)len;             // per-word mean over valid path positions

    if (lane == 0) wavePartial[wave] = part;
    __syncthreads();
    if (threadIdx.x == 0) {
        block_partials[blockIdx.x] =
            wavePartial[0] + wavePartial[1] + wavePartial[2] + wavePartial[3];
    }
}

// Deterministic final reduction: 4096 partials -> scalar mean.
__global__ __launch_bounds__(256) void hs_loss_stage2(
    const float* __restrict__ partials, float* __restrict__ out, int n)
{
    __shared__ float sm[256];
    float s = 0.0f;
    for (int i = threadIdx.x; i < n; i += 256) s += partials[i];
    sm[threadIdx.x] = s;
    __syncthreads();
    #pragma unroll
    for (int stride = 128; stride > 0; stride >>= 1) {
        if (threadIdx.x < stride) sm[threadIdx.x] += sm[threadIdx.x + stride];
        __syncthreads();
    }
    if (threadIdx.x == 0) out[0] = sm[0] / (float)BATCH;
}

extern "C" void kernel_launch(void* const* d_in, const int* in_sizes, int n_in,
                              void* d_out, int out_size, void* d_ws, size_t ws_size,
                              hipStream_t stream) {
    const float* hidden       = (const float*)d_in[0];
    const int*   target_words = (const int*)  d_in[1];
    const float* W            = (const float*)d_in[2];
    const float* path_bits    = (const float*)d_in[3];
    const int*   path_len     = (const int*)  d_in[4];
    float* out = (float*)d_out;
    float* block_partials = (float*)d_ws;    // 4096 floats = 16 KB scratch

    const int nblocks = BATCH / 4;           // one wave per sample, 4 waves/block
    hs_loss_stage1<<<nblocks, 128, 0, stream>>>(hidden, target_words, W,
                                                path_bits, path_len, block_partials);
    hs_loss_stage2<<<1, 256, 0, stream>>>(block_partials, out, nblocks);
}